// VideoRandomPerspective_47330539602571
// MI455X (gfx1250) — compile-verified
//
#include <hip/hip_runtime.h>
#include <hip/hip_bf16.h>
#include <math.h>

// ---------------------------------------------------------------------------
// VideoRandomPerspective — CDNA5 (gfx1250, wave32)
//   prep: noise smoothing + per-frame 3x3 homography composition via
//         V_WMMA_F32_16X16X4_F32 (the only real matrix math in the problem)
//   warp: bandwidth-bound bilinear perspective sampling (NT stores for the
//         streamed 201MB output; gathered input stays resident in 192MB L2)
// ---------------------------------------------------------------------------

typedef __attribute__((ext_vector_type(2))) float v2f;
typedef __attribute__((ext_vector_type(8))) float v8f;

#define L_FRAMES 256   // B*F
#define IMG_H    256
#define IMG_W    256
#define NTAPS    511   // 2*256 - 1
#define NLEN     766   // 3*L - 2

// Solve the 8x8 homography system (f64, partial-pivot Gauss-Jordan) exactly
// as np.linalg.solve does in the reference. i = which corner is tweaked.
__device__ void vrp_solve_basis(int i, float* out9) {
  const double src[4][2] = {{0, 0}, {0, 1}, {1, 0}, {1, 1}};
  const double twk[4][2] = {{-1, 0}, {0, 2}, {1, -1}, {2, 1}};
  double A[8][9];
  for (int p = 0; p < 4; ++p) {
    double sx = ((p == i) ? twk[i][0] : src[p][0]) - 0.5;
    double sy = ((p == i) ? twk[i][1] : src[p][1]) - 0.5;
    double u = src[p][0] - 0.5;
    double v = src[p][1] - 0.5;
    double r0[9] = {sx, sy, 1.0, 0.0, 0.0, 0.0, -u * sx, -u * sy, u};
    double r1[9] = {0.0, 0.0, 0.0, sx, sy, 1.0, -v * sx, -v * sy, v};
    for (int c = 0; c < 9; ++c) { A[2 * p][c] = r0[c]; A[2 * p + 1][c] = r1[c]; }
  }
  for (int col = 0; col < 8; ++col) {
    int piv = col;
    double best = fabs(A[col][col]);
    for (int rr = col + 1; rr < 8; ++rr) {
      double m = fabs(A[rr][col]);
      if (m > best) { best = m; piv = rr; }
    }
    if (piv != col)
      for (int c = col; c < 9; ++c) { double t = A[col][c]; A[col][c] = A[piv][c]; A[piv][c] = t; }
    double inv = 1.0 / A[col][col];
    for (int rr = 0; rr < 8; ++rr) {
      if (rr == col) continue;
      double fq = A[rr][col] * inv;
      for (int c = col; c < 9; ++c) A[rr][c] -= fq * A[col][c];
    }
  }
  for (int k = 0; k < 8; ++k) out9[k] = (float)(A[k][8] / A[k][k]);
  out9[8] = 1.0f;
}

__global__ __launch_bounds__(256) void vrp_prep(const float* __restrict__ noise,
                                                float* __restrict__ Mout) {
  __shared__ float nl[4 * NLEN];      // staged noise
  __shared__ float wl[NTAPS];         // smoothing window
  __shared__ float bas[4][9];         // homography bases
  __shared__ float Wm[L_FRAMES][9];   // per-frame warp matrices
  __shared__ float wsum_sh;

  const int tid = threadIdx.x;

  for (int j = tid; j < 4 * NLEN; j += 256) nl[j] = noise[j];
  for (int j = tid; j < NTAPS; j += 256) {
    int e = (j < 256) ? j : (510 - j);        // palindromic 1.1^min(j,510-j)
    wl[j] = powf(1.1f, (float)e);
  }
  if (tid < 4) vrp_solve_basis(tid, &bas[tid][0]);
  __syncthreads();

  if (tid == 0) {
    float s = 0.0f;
    for (int j = 0; j < NTAPS; ++j) s += wl[j];
    wsum_sh = s;
  }
  __syncthreads();
  {
    float s = wsum_sh;
    for (int j = tid; j < NTAPS; j += 256) wl[j] = wl[j] / s;
  }
  __syncthreads();

  // thread == frame: smoothed (and palindromic => convolution == correlation)
  float aa[4];
#pragma unroll
  for (int c = 0; c < 4; ++c) {
    float acc = 0.0f;
    const float* nr = &nl[c * NLEN + tid];
    for (int k = 0; k < NTAPS; ++k) acc = fmaf(nr[k], wl[k], acc);
    aa[c] = fmaxf(acc, 0.0f);
  }
  float asum = aa[0] + aa[1] + aa[2] + aa[3];
#pragma unroll
  for (int e = 0; e < 9; ++e) {
    float v = aa[0] * bas[0][e] + aa[1] * bas[1][e] + aa[2] * bas[2][e] + aa[3] * bas[3][e];
    if (e == 0 || e == 4 || e == 8) v += (4.0f - asum);
    Wm[tid][e] = v;
  }
  __syncthreads();

  // --- compose M = L * warp * R per frame via V_WMMA_F32_16X16X4_F32 -------
  // L = S1@T1 = [[255,0,127.5],[0,255,127.5],[0,0,1]]
  // R = T2@S2 = [[1/255,0,-0.5],[0,1/255,-0.5],[0,0,1]]
  // A(16x4) layout: lane<16 holds {A[lane][0],A[lane][1]}, lane>=16 {A[l-16][2],A[l-16][3]}
  // B(4x16) layout: v0: lane<16=B[0][lane], lane>=16=B[2][l-16]; v1: B[1][.] / B[3][.]
  // D(16x16) layout: VGPR r, lane<16 -> D[r][lane]
  const int lane = tid & 31;
  const int wavei = tid >> 5;
  const int r = lane & 15;
  const bool hi = (lane >= 16);

  v2f AL; AL.x = 0.0f; AL.y = 0.0f;
  if (!hi) {
    if (r == 0) AL.x = 255.0f;
    else if (r == 1) AL.y = 255.0f;
  } else {
    if (r == 0 || r == 1) AL.x = 127.5f;
    else if (r == 2) AL.x = 1.0f;
  }
  v2f BR; BR.x = 0.0f; BR.y = 0.0f;
  if (!hi) {
    if (lane == 0) BR.x = 1.0f / 255.0f;
    else if (lane == 2) BR.x = -0.5f;
    if (lane == 1) BR.y = 1.0f / 255.0f;
    else if (lane == 2) BR.y = -0.5f;
  } else {
    if (r == 2) BR.x = 1.0f;
  }
  const v8f zacc = {0.f, 0.f, 0.f, 0.f, 0.f, 0.f, 0.f, 0.f};

  for (int i = 0; i < 32; ++i) {                 // wave-uniform: EXEC all-1s at WMMAs
    const int fidx = wavei * 32 + i;
    v2f AW; AW.x = 0.0f; AW.y = 0.0f;
    if (r < 3) {
      if (!hi) { AW.x = Wm[fidx][r * 3 + 0]; AW.y = Wm[fidx][r * 3 + 1]; }
      else     { AW.x = Wm[fidx][r * 3 + 2]; }
    }
    // X = warp * R
    v8f d1 = __builtin_amdgcn_wmma_f32_16x16x4_f32(false, AW, false, BR,
                                                   (short)0, zacc, false, false);
    // re-layout X as a B operand: hi lanes need X row2 from lane (lane-16)
    float x2s = __shfl(d1[2], r, 32);
    v2f BX; BX.x = hi ? x2s : d1[0]; BX.y = hi ? 0.0f : d1[1];
    // M = L * X
    v8f d2 = __builtin_amdgcn_wmma_f32_16x16x4_f32(false, AL, false, BX,
                                                   (short)0, zacc, false, false);
    if (lane < 3) {
      Mout[fidx * 9 + 0 + lane] = d2[0];
      Mout[fidx * 9 + 3 + lane] = d2[1];
      Mout[fidx * 9 + 6 + lane] = d2[2];
    }
  }
}

__global__ __launch_bounds__(256) void vrp_warp(const float* __restrict__ img,
                                                const float* __restrict__ Mws,
                                                float* __restrict__ out) {
  const int bid = blockIdx.x;        // f*H + y
  const int f = bid >> 8;
  const int y = bid & 255;
  const int x = threadIdx.x;

  const float* M = Mws + f * 9;      // block-uniform -> scalar loads
  const float m0 = M[0], m1 = M[1], m2 = M[2];
  const float m3 = M[3], m4 = M[4], m5 = M[5];
  const float m6 = M[6], m7 = M[7], m8 = M[8];

  const float fx = (float)x, fy = (float)y;
  const float u = fmaf(m0, fx, fmaf(m1, fy, m2));
  const float v = fmaf(m3, fx, fmaf(m4, fy, m5));
  const float wq = fmaf(m6, fx, fmaf(m7, fy, m8));
  const float sx = u / wq;
  const float sy = v / wq;

  const float x0f = floorf(sx), y0f = floorf(sy);
  const float wx1 = sx - x0f, wy1 = sy - y0f;
  const float wx0 = 1.0f - wx1, wy0 = 1.0f - wy1;
  const float x1f = x0f + 1.0f, y1f = y0f + 1.0f;

  const float vx0 = (x0f >= 0.0f && x0f <= 255.0f) ? 1.0f : 0.0f;
  const float vx1 = (x1f >= 0.0f && x1f <= 255.0f) ? 1.0f : 0.0f;
  const float vy0 = (y0f >= 0.0f && y0f <= 255.0f) ? 1.0f : 0.0f;
  const float vy1 = (y1f >= 0.0f && y1f <= 255.0f) ? 1.0f : 0.0f;

  const int ix0 = (int)fminf(fmaxf(x0f, 0.0f), 255.0f);
  const int ix1 = (int)fminf(fmaxf(x1f, 0.0f), 255.0f);
  const int iy0 = (int)fminf(fmaxf(y0f, 0.0f), 255.0f);
  const int iy1 = (int)fminf(fmaxf(y1f, 0.0f), 255.0f);

  const float w00 = wx0 * wy0 * vx0 * vy0;
  const float w10 = wx1 * wy0 * vx1 * vy0;
  const float w01 = wx0 * wy1 * vx0 * vy1;
  const float w11 = wx1 * wy1 * vx1 * vy1;

  const float* base = img + (size_t)f * (IMG_H * IMG_W * 3);
  const float* p00 = base + (iy0 * IMG_W + ix0) * 3;
  const float* p10 = base + (iy0 * IMG_W + ix1) * 3;
  const float* p01 = base + (iy1 * IMG_W + ix0) * 3;
  const float* p11 = base + (iy1 * IMG_W + ix1) * 3;

  const float cr = w00 * p00[0] + w10 * p10[0] + w01 * p01[0] + w11 * p11[0];
  const float cg = w00 * p00[1] + w10 * p10[1] + w01 * p01[1] + w11 * p11[1];
  const float cb = w00 * p00[2] + w10 * p10[2] + w01 * p01[2] + w11 * p11[2];

  float* op = out + ((size_t)bid * IMG_W + x) * 3;
  __builtin_nontemporal_store(cr, op + 0);   // stream output, keep L2 for gathers
  __builtin_nontemporal_store(cg, op + 1);
  __builtin_nontemporal_store(cb, op + 2);
}

extern "C" void kernel_launch(void* const* d_in, const int* in_sizes, int n_in,
                              void* d_out, int out_size, void* d_ws, size_t ws_size,
                              hipStream_t stream) {
  const float* x = (const float*)d_in[0];       // (B,F,H,W,C) f32
  const float* noise = (const float*)d_in[1];   // (4, 3*l-2) f32
  float* Mws = (float*)d_ws;                    // 256*9 floats of per-frame M

  const int l = in_sizes[0] / (IMG_H * IMG_W * 3);   // = 256

  vrp_prep<<<1, 256, 0, stream>>>(noise, Mws);
  vrp_warp<<<l * IMG_H, 256, 0, stream>>>(x, Mws, (float*)d_out);
}